// CommutatorConv2d_76587856822725
// MI455X (gfx1250) — compile-verified
//
#include <hip/hip_runtime.h>

typedef __attribute__((ext_vector_type(2))) float v2f;
typedef __attribute__((ext_vector_type(8))) float v8f;

namespace {
constexpr int B  = 16;
constexpr int C  = 32;
constexpr int H  = 128;
constexpr int W  = 128;
constexpr int HW = H * W;  // 16384 = channel stride
}

// ---------------------------------------------------------------------------
// Kernel 1: xs[b,y,x] = sum_c x[b,c,y,x]   via V_WMMA_F32_16X16X4_F32.
//
// A = all-ones 16x4  =>  D[m,n] = sum_k B[k,n]  (every row of D identical),
// so the result does not depend on the exact K <-> (VGPR, lane-half) mapping
// of the B operand -- only that the 4 slots hold 4 distinct channels.
//
// Each wave: 16 pixels, 8 accumulating WMMAs x 4 channels = 32 channels.
// Block = 8 waves = one full 128-wide image row (fully coalesced loads).
// ---------------------------------------------------------------------------
__global__ __launch_bounds__(256) void chansum_wmma_kernel(
    const float* __restrict__ x, float* __restrict__ xs) {
  const int row  = blockIdx.x;          // 0 .. B*H-1
  const int b    = row >> 7;            // row / 128
  const int y    = row & 127;
  const int lane = threadIdx.x & 31;
  const int wave = threadIdx.x >> 5;    // 0..7
  const int col  = (wave << 4) | (lane & 15);
  const int half = lane >> 4;           // 0 or 1

  const float* base = x + (size_t)b * C * HW + (size_t)y * W + col;

  v2f a;
  a[0] = 1.0f;
  a[1] = 1.0f;
  v8f acc = {};
#pragma unroll
  for (int c0 = 0; c0 < C; c0 += 4) {
    const int c = c0 + 2 * half;        // this lane-half supplies 2 channels
    v2f bm;
    bm[0] = base[(size_t)c * HW];
    bm[1] = base[(size_t)(c + 1) * HW];
    // D = A(ones) x B + C  ->  per-column channel partial sums, accumulated.
    acc = __builtin_amdgcn_wmma_f32_16x16x4_f32(false, a, false, bm, (short)0,
                                                acc, false, false);
  }
  // All 16 rows of D are identical; lanes 16..31 duplicate columns of 0..15.
  if (lane < 16) {
    xs[(size_t)b * HW + (size_t)y * W + col] = acc[0];
  }
}

// ---------------------------------------------------------------------------
// Kernel 2: out[b,y,x] = sum_{i,j} xs_pad[b, y+i-1, x+j-1] * W[i][j] + bias
// with W[i][j] = (lc+la)*colsumK[j] + (la-lc)*rowsumK[i].
// xs (1 MiB) sits in L2 between kernels; this pass is latency-trivial.
// ---------------------------------------------------------------------------
__global__ __launch_bounds__(256) void conv3x3_kernel(
    const float* __restrict__ xs, const float* __restrict__ Kf,
    const float* __restrict__ bias, const float* __restrict__ lc,
    const float* __restrict__ la, float* __restrict__ out) {
  const int idx = blockIdx.x * 256 + threadIdx.x;  // 0 .. B*H*W-1
  const int b   = idx >> 14;                       // / 16384
  const int p   = idx & (HW - 1);
  const int y   = p >> 7;
  const int xc  = p & 127;

  const float lam_c = lc[0];
  const float lam_a = la[0];
  const float wa = lam_c + lam_a;  // weight on column sums of K
  const float wb = lam_a - lam_c;  // weight on row sums of K

  float cK[3], rK[3];
#pragma unroll
  for (int j = 0; j < 3; ++j) cK[j] = Kf[j] + Kf[3 + j] + Kf[6 + j];
#pragma unroll
  for (int i = 0; i < 3; ++i) rK[i] = Kf[3 * i] + Kf[3 * i + 1] + Kf[3 * i + 2];

  const float* s = xs + (size_t)b * HW;
  float acc = 0.0f;
#pragma unroll
  for (int i = 0; i < 3; ++i) {
    const int yy = y + i - 1;
#pragma unroll
    for (int j = 0; j < 3; ++j) {
      const int xx = xc + j - 1;
      const bool inb = (yy >= 0) & (yy < H) & (xx >= 0) & (xx < W);
      const float v = inb ? s[yy * W + xx] : 0.0f;
      acc = fmaf(v, wa * cK[j] + wb * rK[i], acc);
    }
  }
  out[idx] = acc + bias[0];
}

extern "C" void kernel_launch(void* const* d_in, const int* in_sizes, int n_in,
                              void* d_out, int out_size, void* d_ws,
                              size_t ws_size, hipStream_t stream) {
  // setup_inputs order: x, K, bias, lambda_c, lambda_a  (all float32)
  const float* x    = (const float*)d_in[0];
  const float* Kf   = (const float*)d_in[1];
  const float* bias = (const float*)d_in[2];
  const float* lc   = (const float*)d_in[3];
  const float* la   = (const float*)d_in[4];
  float* xs  = (float*)d_ws;   // B*H*W floats = 1 MiB scratch
  float* out = (float*)d_out;  // B*1*H*W floats

  chansum_wmma_kernel<<<B * H, 256, 0, stream>>>(x, xs);
  conv3x3_kernel<<<(B * HW) / 256, 256, 0, stream>>>(xs, Kf, bias, lc, la, out);
}